// AttnDecoder_42228118454895
// MI455X (gfx1250) — compile-verified
//
#include <hip/hip_runtime.h>
#include <hip/hip_bf16.h>
#include <math.h>

typedef __attribute__((ext_vector_type(2))) float v2f;
typedef __attribute__((ext_vector_type(8))) float v8f;

#define HDIM 1024
#define EDIM 1024
#define LDIM 2048
#define VDIM 50257
#define XDIM 2048 /* H + E */

__device__ __forceinline__ float waveSum(float v) {
#pragma unroll
  for (int o = 16; o > 0; o >>= 1) v += __shfl_down(v, o, 32);
  return v;
}

// K1: d[e] = W_t[e,:]·h_i + b_t[e] + emb[y,e]; also xbuf[e] = emb[y,e]
__global__ void k_embed_lin(const int* __restrict__ y_i, const float* __restrict__ h_i,
                            const float* __restrict__ emb, const float* __restrict__ W_t,
                            const float* __restrict__ b_t, float* __restrict__ dvec,
                            float* __restrict__ xbuf) {
  int wave = threadIdx.x >> 5, lane = threadIdx.x & 31;
  int e = blockIdx.x * 8 + wave;
  const float4* wr = (const float4*)(W_t + (size_t)e * HDIM);
  const float4* hv = (const float4*)h_i;
  float s = 0.f;
#pragma unroll 4
  for (int i = lane; i < HDIM / 4; i += 32) {
    float4 w = wr[i], x = hv[i];
    s += w.x * x.x + w.y * x.y + w.z * x.z + w.w * x.w;
  }
  s = waveSum(s);
  if (lane == 0) {
    int y = y_i[0];
    float g = emb[(size_t)y * HDIM + e];
    dvec[e] = s + b_t[e] + g;
    xbuf[e] = g;
  }
}

// K2: scores[l] = sum_e d[e] * cnn_a[e*L + l]
__global__ void k_scores(const float* __restrict__ dvec, const float* __restrict__ cnn_a,
                         float* __restrict__ scores) {
  __shared__ float sd[EDIM];
  __shared__ float acc[8][32];
  int tid = threadIdx.x;
  for (int i = tid; i < EDIM; i += 256) sd[i] = dvec[i];
  __syncthreads();
  int lsub = tid & 31, esl = tid >> 5;
  int l = blockIdx.x * 32 + lsub;
  float s = 0.f;
  int e0 = esl * (EDIM / 8);
#pragma unroll 4
  for (int e = e0; e < e0 + EDIM / 8; ++e) s += sd[e] * cnn_a[(size_t)e * LDIM + l];
  acc[esl][lsub] = s;
  __syncthreads();
  if (esl == 0) {
    float t = 0.f;
#pragma unroll
    for (int j = 0; j < 8; ++j) t += acc[j][lsub];
    scores[l] = t;
  }
}

// K3: attn = softmax(scores) over L=2048, single block of 1024
__global__ void k_softmax(const float* __restrict__ scores, float* __restrict__ attn) {
  __shared__ float red[1024];
  int tid = threadIdx.x;
  float v0 = scores[tid], v1 = scores[tid + 1024];
  red[tid] = fmaxf(v0, v1);
  __syncthreads();
  for (int s = 512; s > 0; s >>= 1) {
    if (tid < s) red[tid] = fmaxf(red[tid], red[tid + s]);
    __syncthreads();
  }
  float m = red[0];
  __syncthreads();
  float e0 = expf(v0 - m), e1 = expf(v1 - m);
  red[tid] = e0 + e1;
  __syncthreads();
  for (int s = 512; s > 0; s >>= 1) {
    if (tid < s) red[tid] += red[tid + s];
    __syncthreads();
  }
  float inv = 1.0f / red[0];
  attn[tid] = e0 * inv;
  attn[tid + 1024] = e1 * inv;
}

// K4: xbuf[H + h] = c[h] = sum_l attn[l] * cnn_c[h*L + l]
__global__ void k_context(const float* __restrict__ attn, const float* __restrict__ cnn_c,
                          float* __restrict__ xbuf) {
  int wave = threadIdx.x >> 5, lane = threadIdx.x & 31;
  int h = blockIdx.x * 8 + wave;
  const float4* cr = (const float4*)(cnn_c + (size_t)h * LDIM);
  const float4* av = (const float4*)attn;
  float s = 0.f;
#pragma unroll 4
  for (int i = lane; i < LDIM / 4; i += 32) {
    float4 c = cr[i], a = av[i];
    s += c.x * a.x + c.y * a.y + c.z * a.z + c.w * a.w;
  }
  s = waveSum(s);
  if (lane == 0) xbuf[HDIM + h] = s;
}

// K5: fused single-step GRU (PyTorch gate order r,z,n), wave per output j
__global__ void k_gru(const float* __restrict__ xbuf, const float* __restrict__ h_i,
                      const float* __restrict__ W_ih, const float* __restrict__ W_hh,
                      const float* __restrict__ b_ih, const float* __restrict__ b_hh,
                      float* __restrict__ hnew) {
  int wave = threadIdx.x >> 5, lane = threadIdx.x & 31;
  int j = blockIdx.x * 8 + wave;
  const float4* xv = (const float4*)xbuf;
  const float4* hv = (const float4*)h_i;
  const float4* wr = (const float4*)(W_ih + (size_t)j * XDIM);
  const float4* wz = (const float4*)(W_ih + (size_t)(HDIM + j) * XDIM);
  const float4* wn = (const float4*)(W_ih + (size_t)(2 * HDIM + j) * XDIM);
  const float4* ur = (const float4*)(W_hh + (size_t)j * HDIM);
  const float4* uz = (const float4*)(W_hh + (size_t)(HDIM + j) * HDIM);
  const float4* un = (const float4*)(W_hh + (size_t)(2 * HDIM + j) * HDIM);
  float sr = 0, sz = 0, sn = 0, tr = 0, tz = 0, tn = 0;
#pragma unroll 2
  for (int i = lane; i < XDIM / 4; i += 32) {
    float4 x = xv[i];
    float4 a = wr[i]; sr += a.x * x.x + a.y * x.y + a.z * x.z + a.w * x.w;
    float4 b = wz[i]; sz += b.x * x.x + b.y * x.y + b.z * x.z + b.w * x.w;
    float4 c = wn[i]; sn += c.x * x.x + c.y * x.y + c.z * x.z + c.w * x.w;
  }
#pragma unroll 2
  for (int i = lane; i < HDIM / 4; i += 32) {
    float4 x = hv[i];
    float4 a = ur[i]; tr += a.x * x.x + a.y * x.y + a.z * x.z + a.w * x.w;
    float4 b = uz[i]; tz += b.x * x.x + b.y * x.y + b.z * x.z + b.w * x.w;
    float4 c = un[i]; tn += c.x * x.x + c.y * x.y + c.z * x.z + c.w * x.w;
  }
  sr = waveSum(sr); sz = waveSum(sz); sn = waveSum(sn);
  tr = waveSum(tr); tz = waveSum(tz); tn = waveSum(tn);
  if (lane == 0) {
    float gr = sr + b_ih[j] + tr + b_hh[j];
    float gz = sz + b_ih[HDIM + j] + tz + b_hh[HDIM + j];
    float r = 1.0f / (1.0f + expf(-gr));
    float z = 1.0f / (1.0f + expf(-gz));
    float n = tanhf(sn + b_ih[2 * HDIM + j] + r * (tn + b_hh[2 * HDIM + j]));
    hnew[j] = (1.0f - z) * n + z * h_i[j];
  }
}

// K6: logits via WMMA f32 16x16x4. Wave owns 16 rows of W_o (A matrix),
// B = h chunk broadcast across all 16 columns -> D[m][*] = W_o[m,:]·h.
// Also emits per-block online-softmax partials (max, sum exp).
__global__ void k_logits(const float* __restrict__ Wo, const float* __restrict__ bo,
                         const float* __restrict__ hn, float* __restrict__ logits,
                         float* __restrict__ pmax, float* __restrict__ psum) {
  __shared__ float sh[HDIM];
  __shared__ float lvals[128];
  __shared__ float red[128];
  int tid = threadIdx.x;
  for (int i = tid; i < HDIM; i += 256) sh[i] = hn[i];
  __syncthreads();

  int wave = tid >> 5, lane = tid & 31;
  int col = lane & 15, half = lane >> 4;
  int vbase = blockIdx.x * 128 + wave * 16;
  int vrow = vbase + col;                       // A-matrix row this lane feeds
  size_t vcl = (vrow < VDIM) ? (size_t)vrow : (size_t)(VDIM - 1);
  const v2f* wr = (const v2f*)(Wo + vcl * HDIM); // lane half selects K pair {0,1} vs {2,3}
  const v2f* hv = (const v2f*)sh;

  v8f acc = {0.f, 0.f, 0.f, 0.f, 0.f, 0.f, 0.f, 0.f};
  for (int ko = 0; ko < 32; ++ko) {             // 32 * 8 = 256 WMMA steps, K=4 each
    if (ko < 16) __builtin_prefetch((const void*)(wr + 2 * ko * 8 + 256), 0, 0);
#pragma unroll
    for (int ki = 0; ki < 8; ++ki) {
      int k4 = ko * 8 + ki;
      v2f a = wr[2 * k4 + half];                // A: rows m, K = 4*k4 + 2*half .. +1
      v2f b = hv[2 * k4 + half];                // B: h[K], h[K+1] broadcast to all N
      acc = __builtin_amdgcn_wmma_f32_16x16x4_f32(
          /*neg_a=*/false, a, /*neg_b=*/false, b,
          /*c_mod=*/(short)0, acc, /*reuse_a=*/false, /*reuse_b=*/false);
    }
  }

  // D layout: VGPR r holds M=r (lanes 0-15) and M=8+r (lanes 16-31); all N equal.
#pragma unroll
  for (int r = 0; r < 8; ++r) {
    if (col == r) {
      int row = r + 8 * half;
      int v = vbase + row;
      float val;
      if (v < VDIM) {
        val = acc[r] + bo[v];
        logits[v] = val;
      } else {
        val = -INFINITY;
      }
      lvals[wave * 16 + row] = val;
    }
  }
  __syncthreads();

  // Block online-softmax partial over its 128 rows
  if (tid < 128) red[tid] = lvals[tid];
  __syncthreads();
  for (int s = 64; s > 0; s >>= 1) {
    if (tid < s) red[tid] = fmaxf(red[tid], red[tid + s]);
    __syncthreads();
  }
  float M = red[0];
  __syncthreads();
  if (tid < 128) red[tid] = (lvals[tid] == -INFINITY) ? 0.0f : expf(lvals[tid] - M);
  __syncthreads();
  for (int s = 64; s > 0; s >>= 1) {
    if (tid < s) red[tid] += red[tid + s];
    __syncthreads();
  }
  if (tid == 0) {
    pmax[blockIdx.x] = M;
    psum[blockIdx.x] = red[0];
  }
}

// K7: combine per-block (max, sumexp) partials -> global max and log-sum
__global__ void k_combine(const float* __restrict__ pmax, const float* __restrict__ psum,
                          float* __restrict__ gstats, int nb) {
  __shared__ float red[512];
  int tid = threadIdx.x;
  float m = -INFINITY;
  for (int i = tid; i < nb; i += 512) m = fmaxf(m, pmax[i]);
  red[tid] = m;
  __syncthreads();
  for (int s = 256; s > 0; s >>= 1) {
    if (tid < s) red[tid] = fmaxf(red[tid], red[tid + s]);
    __syncthreads();
  }
  float M = red[0];
  __syncthreads();
  float s = 0.f;
  for (int i = tid; i < nb; i += 512) s += psum[i] * expf(pmax[i] - M);
  red[tid] = s;
  __syncthreads();
  for (int st = 256; st > 0; st >>= 1) {
    if (tid < st) red[tid] += red[tid + st];
    __syncthreads();
  }
  if (tid == 0) {
    gstats[0] = M;
    gstats[1] = logf(red[0]);
  }
}

// K8: out[0:V] = log_softmax(logits), out[V:V+H] = gru_hidden
__global__ void k_final(const float* __restrict__ logits, const float* __restrict__ hnew,
                        const float* __restrict__ gstats, float* __restrict__ out) {
  int i = blockIdx.x * 256 + threadIdx.x;
  float M = gstats[0], LS = gstats[1];
  if (i < VDIM) out[i] = logits[i] - M - LS;
  else if (i < VDIM + HDIM) out[i] = hnew[i - VDIM];
}

extern "C" void kernel_launch(void* const* d_in, const int* in_sizes, int n_in,
                              void* d_out, int out_size, void* d_ws, size_t ws_size,
                              hipStream_t stream) {
  (void)in_sizes; (void)n_in; (void)out_size; (void)ws_size;
  const int*   y_i   = (const int*)d_in[0];
  const float* h_i   = (const float*)d_in[1];
  const float* cnn_a = (const float*)d_in[2];
  const float* cnn_c = (const float*)d_in[3];
  const float* emb   = (const float*)d_in[4];
  const float* W_t   = (const float*)d_in[5];
  const float* b_t   = (const float*)d_in[6];
  const float* W_ih  = (const float*)d_in[7];
  const float* W_hh  = (const float*)d_in[8];
  const float* b_ih  = (const float*)d_in[9];
  const float* b_hh  = (const float*)d_in[10];
  const float* W_o   = (const float*)d_in[11];
  const float* b_o   = (const float*)d_in[12];

  float* ws     = (float*)d_ws;
  float* dvec   = ws;             // 1024
  float* scores = ws + 1024;      // 2048
  float* attn   = ws + 3072;      // 2048
  float* xbuf   = ws + 5120;      // 2048 = [g | c]
  float* hnew   = ws + 7168;      // 1024
  float* logits = ws + 8192;      // 50304 (padded)
  float* pmax   = ws + 58624;     // 512
  float* psum   = ws + 59136;     // 512
  float* gstats = ws + 59648;     // 2
  float* out    = (float*)d_out;

  k_embed_lin<<<EDIM / 8, 256, 0, stream>>>(y_i, h_i, emb, W_t, b_t, dvec, xbuf);
  k_scores<<<LDIM / 32, 256, 0, stream>>>(dvec, cnn_a, scores);
  k_softmax<<<1, 1024, 0, stream>>>(scores, attn);
  k_context<<<HDIM / 8, 256, 0, stream>>>(attn, cnn_c, xbuf);
  k_gru<<<HDIM / 8, 256, 0, stream>>>(xbuf, h_i, W_ih, W_hh, b_ih, b_hh, hnew);
  int nb = (VDIM + 127) / 128;  // 393 blocks, 16 rows per wave
  k_logits<<<nb, 256, 0, stream>>>(W_o, b_o, hnew, logits, pmax, psum);
  k_combine<<<1, 512, 0, stream>>>(pmax, psum, gstats, nb);
  int nf = (VDIM + HDIM + 255) / 256;
  k_final<<<nf, 256, 0, stream>>>(logits, hnew, gstats, out);
}